// MultiHeadAttention_5497558139211
// MI455X (gfx1250) — compile-verified
//
#include <hip/hip_runtime.h>
#include <hip/hip_bf16.h>
#include <math.h>

typedef __attribute__((ext_vector_type(2))) float v2f;
typedef __attribute__((ext_vector_type(4))) float v4f;
typedef __attribute__((ext_vector_type(8))) float v8f;

#define B_    2
#define T_    2048
#define D_    1024
#define H_    8
#define HD_   128
#define CLIP_ 2
#define MROWS (B_ * T_)   // 4096

// Exact-precision fp32 WMMA: D = A(16x4) * B(4x16) + C(16x16)
static __device__ __forceinline__ v8f wmma_f32(v2f a, v2f b, v8f c) {
  return __builtin_amdgcn_wmma_f32_16x16x4_f32(false, a, false, b, (short)0, c,
                                               false, false);
}

// CDNA5 async copy: global -> LDS, 16B per lane, tracked by ASYNCcnt.
static __device__ __forceinline__ void async_copy_b128(uint32_t lds_byte_addr,
                                                       const float* g) {
  asm volatile("global_load_async_to_lds_b128 %0, %1, off"
               :
               : "v"(lds_byte_addr), "v"((unsigned long long)(size_t)g)
               : "memory");
}
static __device__ __forceinline__ void wait_asynccnt0() {
  asm volatile("s_wait_asynccnt 0" ::: "memory");
}
static __device__ __forceinline__ uint32_t lds_addr(const void* p) {
  // flat address of LDS: low 32 bits are the workgroup-relative LDS offset
  return (uint32_t)(size_t)p;
}

// -------------------------------------------------------------------------
// Y[M][N] = X[M][K] @ W[N][K]^T + bias[N]
// 256 threads (8 waves). Block tile BM=32 x BN=128, BK=32, double-buffered
// LDS fed by global_load_async_to_lds_b128. Wave (wr,wc) computes 16x32.
// grid = (N/128, M/32)
// -------------------------------------------------------------------------
#define BM 32
#define BN 128
#define BK 32
#define LP 36   // LDS row pitch (floats): 144B, 16B-aligned, conflict-free

__global__ __launch_bounds__(256) void gemm_async(
    const float* __restrict__ X, const float* __restrict__ W,
    const float* __restrict__ bias, float* __restrict__ Y, int M, int N,
    int K) {
  __shared__ float Xs[2][BM][LP];
  __shared__ float Ws[2][BN][LP];

  const int tid = threadIdx.x;
  const int lane = tid & 31;
  const int wave = tid >> 5;
  const int hl = lane >> 4;
  const int ln = lane & 15;
  const int kL = hl * 2;
  const int wr = wave >> 2;  // m-subtile (0..1)
  const int wc = wave & 3;   // n-group of 32 (0..3)
  const int m0 = blockIdx.y * BM;
  const int n0 = blockIdx.x * BN;

  // loader mapping: 8 threads per row, 16B per thread, 32 rows per pass
  const int lrow = tid >> 3;
  const int lcol = (tid & 7) * 4;
  const float* xg = X + (size_t)(m0 + lrow) * K + lcol;
  const float* wg = W + (size_t)(n0 + lrow) * K + lcol;

  const uint32_t xl0 = lds_addr(&Xs[0][lrow][lcol]);
  const uint32_t wl0 = lds_addr(&Ws[0][lrow][lcol]);
  const uint32_t XBUF = (uint32_t)(BM * LP * 4);
  const uint32_t WBUF = (uint32_t)(BN * LP * 4);

  auto issue = [&](int buf, int kb) {
    async_copy_b128(xl0 + (uint32_t)buf * XBUF, xg + kb);
#pragma unroll
    for (int i = 0; i < 4; ++i)
      async_copy_b128(wl0 + (uint32_t)buf * WBUF + (uint32_t)(32 * LP * 4) * i,
                      wg + (size_t)(32 * i) * K + kb);
  };

  v8f acc0 = {}, acc1 = {};
  issue(0, 0);
  wait_asynccnt0();
  __syncthreads();

  const int nk = K / BK;
  for (int kc = 0; kc < nk; ++kc) {
    const int cur = kc & 1;
    if (kc + 1 < nk) issue(cur ^ 1, (kc + 1) * BK);

    const float* xa = &Xs[cur][wr * 16 + ln][kL];
    const float* wb0 = &Ws[cur][wc * 32 + ln][kL];
    const float* wb1 = &Ws[cur][wc * 32 + 16 + ln][kL];
#pragma unroll
    for (int ks = 0; ks < 8; ++ks) {
      const v2f a = *(const v2f*)(xa + ks * 4);
      acc0 = wmma_f32(a, *(const v2f*)(wb0 + ks * 4), acc0);
      acc1 = wmma_f32(a, *(const v2f*)(wb1 + ks * 4), acc1);
    }

    if (kc + 1 < nk) wait_asynccnt0();
    __syncthreads();
  }

  const int nc0 = n0 + wc * 32 + ln;
  const float bv0 = bias[nc0];
  const float bv1 = bias[nc0 + 16];
#pragma unroll
  for (int r = 0; r < 8; ++r) {
    const size_t row = (size_t)(m0 + wr * 16 + r + 8 * hl);
    Y[row * N + nc0] = acc0[r] + bv0;
    Y[row * N + nc0 + 16] = acc1[r] + bv1;
  }
}

// -------------------------------------------------------------------------
// proj[b][t2][r] = sum_k tq[b][t2][k] * table[r][k]   (r = 0..4)
// -------------------------------------------------------------------------
__global__ __launch_bounds__(256) void relproj_kernel(
    const float* __restrict__ tq, const float* __restrict__ table,
    float* __restrict__ proj) {
  __shared__ float red[5][256];
  const int row = blockIdx.x;  // b*T_ + t2
  const int tid = threadIdx.x;
  const float* x = tq + (size_t)row * D_;
  float a0 = 0.f, a1 = 0.f, a2 = 0.f, a3 = 0.f, a4 = 0.f;
  for (int k = tid; k < D_; k += 256) {
    const float xv = x[k];
    a0 += xv * table[0 * D_ + k];
    a1 += xv * table[1 * D_ + k];
    a2 += xv * table[2 * D_ + k];
    a3 += xv * table[3 * D_ + k];
    a4 += xv * table[4 * D_ + k];
  }
  red[0][tid] = a0; red[1][tid] = a1; red[2][tid] = a2;
  red[3][tid] = a3; red[4][tid] = a4;
  __syncthreads();
  for (int s = 128; s > 0; s >>= 1) {
    if (tid < s) {
#pragma unroll
      for (int r = 0; r < 5; ++r) red[r][tid] += red[r][tid + s];
    }
    __syncthreads();
  }
  if (tid < 5) proj[(size_t)row * 5 + tid] = red[tid][0];
}

// -------------------------------------------------------------------------
// Pass A: logits[b][h][t2][t1] = (q.k + rel) / sqrt(HD); online row max/sum.
// 256 thr (8 waves) per block; block owns a 64-row q tile.
// Waves: wr = wave>>1 (m-subtile 0..3), wc = wave&1 (n-tile 0..1).
// K-chunks of 32 t1-rows double-buffered via async copies.
// grid = B*H*(T/64)
// -------------------------------------------------------------------------
#define KCH 32
#define KP  132  // pitch for 128-float rows: 528B, 16B-aligned, banks 4r%64

__global__ __launch_bounds__(256) void attn_logits(
    const float* __restrict__ tq, const float* __restrict__ tk,
    const float* __restrict__ proj, float* __restrict__ logits,
    float* __restrict__ rowmax, float* __restrict__ rowsum) {
  __shared__ float Ks[2][KCH][KP];
  const int tid = threadIdx.x;
  const int lane = tid & 31;
  const int wave = tid >> 5;
  const int hl = lane >> 4, ln = lane & 15, kL = hl * 2;
  const int wr = wave >> 1;  // 0..3
  const int wc = wave & 1;   // 0..1
  int blk = blockIdx.x;
  const int qt = blk & (T_ / 64 - 1); blk >>= 5;
  const int h = blk & (H_ - 1);       blk >>= 3;
  const int b = blk;
  const int qbase = qt * 64;

  // preload this wave's 16-row Q sub-tile fragments (64 VGPRs)
  v2f qa[32];
  const float* qrow =
      tq + ((size_t)b * T_ + qbase + wr * 16 + ln) * D_ + h * HD_ + kL;
#pragma unroll
  for (int ks = 0; ks < 32; ++ks) qa[ks] = *(const v2f*)(qrow + ks * 4);

  // loader mapping: 32 rows x 32 chunks of 16B; 4 copies per thread
  const float* khead = tk + (size_t)b * T_ * D_ + h * HD_;
  const int lc = (tid & 31) * 4;
  const int lr = tid >> 5;  // + 8*i
  const uint32_t kbase_l = lds_addr(&Ks[0][lr][lc]);
  const uint32_t KBUF = (uint32_t)(KCH * KP * 4);

  auto issue = [&](int buf, int t10) {
#pragma unroll
    for (int i = 0; i < 4; ++i)
      async_copy_b128(kbase_l + (uint32_t)buf * KBUF +
                          (uint32_t)(8 * KP * 4) * i,
                      khead + (size_t)(t10 + lr + 8 * i) * D_ + lc);
  };

  const float scale = 0.088388347648318447f;  // 1/sqrt(128)
  float rm[8], rs[8];
#pragma unroll
  for (int r = 0; r < 8; ++r) { rm[r] = -INFINITY; rs[r] = 0.f; }

  float* lbase = logits + ((size_t)b * H_ + h) * T_ * T_;

  issue(0, 0);
  wait_asynccnt0();
  __syncthreads();

  const int nch = T_ / KCH;  // 64
  for (int ch = 0; ch < nch; ++ch) {
    const int cur = ch & 1;
    if (ch + 1 < nch) issue(cur ^ 1, (ch + 1) * KCH);

    const float* kb_ = &Ks[cur][wc * 16 + ln][kL];
    v8f acc = {};
#pragma unroll 8
    for (int ks = 0; ks < 32; ++ks)
      acc = wmma_f32(qa[ks], *(const v2f*)(kb_ + ks * 4), acc);

    const int t1 = ch * KCH + wc * 16 + ln;
#pragma unroll
    for (int r = 0; r < 8; ++r) {
      const int t2 = qbase + wr * 16 + r + 8 * hl;
      int dp = t1 - t2;
      dp = dp < -CLIP_ ? -CLIP_ : (dp > CLIP_ ? CLIP_ : dp);
      const float rel = proj[((size_t)b * T_ + t2) * 5 + dp + CLIP_];
      const float x = (acc[r] + rel) * scale;
      lbase[(size_t)t2 * T_ + t1] = x;
      const float nm = fmaxf(rm[r], x);
      rs[r] = rs[r] * __expf(rm[r] - nm) + __expf(x - nm);
      rm[r] = nm;
    }

    if (ch + 1 < nch) wait_asynccnt0();
    __syncthreads();
  }

  // reduce (m,s) across the 16 lanes of each half-wave
#pragma unroll
  for (int r = 0; r < 8; ++r) {
#pragma unroll
    for (int off = 8; off >= 1; off >>= 1) {
      const float om = __shfl_xor(rm[r], off, 32);
      const float os = __shfl_xor(rs[r], off, 32);
      const float nm = fmaxf(rm[r], om);
      rs[r] = rs[r] * __expf(rm[r] - nm) + os * __expf(om - nm);
      rm[r] = nm;
    }
  }

  __shared__ float sm[8][16], ss[8][16];
  if (ln == 0) {
#pragma unroll
    for (int r = 0; r < 8; ++r) {
      sm[wave][hl * 8 + r] = rm[r];
      ss[wave][hl * 8 + r] = rs[r];
    }
  }
  __syncthreads();
  if (tid < 64) {
    const int msub = tid >> 4;    // which m-subtile
    const int rloc = tid & 15;    // row within subtile
    const int w0 = msub * 2, w1 = msub * 2 + 1;  // the two waves (wc=0,1)
    float m = sm[w0][rloc], s = ss[w0][rloc];
    const float om = sm[w1][rloc], os = ss[w1][rloc];
    const float nm = fmaxf(m, om);
    s = s * __expf(m - nm) + os * __expf(om - nm);
    m = nm;
    const size_t rb = ((size_t)b * H_ + h) * T_ + qbase + tid;
    rowmax[rb] = m;
    rowsum[rb] = s;
  }
}

// -------------------------------------------------------------------------
// Pass B: weights = exp(x - rowmax) / rowsum, in place, float4 vectorized
// -------------------------------------------------------------------------
__global__ __launch_bounds__(256) void softmax_norm(
    float* __restrict__ w, const float* __restrict__ rowmax,
    const float* __restrict__ rowsum) {
  const size_t i4 = (size_t)blockIdx.x * blockDim.x + threadIdx.x;
  const size_t base = i4 * 4;
  const size_t row = base / T_;
  const float m = rowmax[row];
  const float inv = 1.0f / rowsum[row];
  v4f x = *(v4f*)(w + base);
  x.x = __expf(x.x - m) * inv;
  x.y = __expf(x.y - m) * inv;
  x.z = __expf(x.z - m) * inv;
  x.w = __expf(x.w - m) * inv;
  *(v4f*)(w + base) = x;
}

// -------------------------------------------------------------------------
// Pass C: ctx[b][t2][h*HD+n] = sum_t1 weights[b][h][t2][t1] * tv[b][t1][h*HD+n]
// 256 thr (8 waves); block owns 32 m-rows x full 128-col head.
// Waves: wr = wave>>2 (m-subtile 0..1), wc = wave&3 (32-col group).
// K-chunks of 32 t1-rows; weights tile + V tile async double-buffered.
// grid = B*H*(T/32)
// -------------------------------------------------------------------------
#define AP 36   // pitch for 32-float weight-tile rows

__global__ __launch_bounds__(256) void attn_ctx(
    const float* __restrict__ w, const float* __restrict__ tv,
    float* __restrict__ ctx) {
  __shared__ float At[2][32][AP];
  __shared__ float Vt[2][KCH][KP];
  const int tid = threadIdx.x;
  const int lane = tid & 31;
  const int wave = tid >> 5;
  const int hl = lane >> 4, ln = lane & 15, kL = hl * 2;
  const int wr = wave >> 2;  // 0..1
  const int wc = wave & 3;   // 0..3
  int blk = blockIdx.x;
  const int mt = blk & (T_ / 32 - 1); blk >>= 6;
  const int h = blk & (H_ - 1);       blk >>= 3;
  const int b = blk;
  const int m0 = mt * 32;

  // loader mapping (weights A tile: 32 rows x 8 chunks -> 1 per thread)
  const float* wrowg = w + (((size_t)b * H_ + h) * T_ + m0) * T_;
  const int ar = tid >> 3;
  const int ac = (tid & 7) * 4;
  const uint32_t abase = lds_addr(&At[0][ar][ac]);
  const uint32_t ABUF = (uint32_t)(32 * AP * 4);

  // loader mapping (V tile: 32 rows x 32 chunks -> 4 per thread)
  const float* vhead = tv + (size_t)b * T_ * D_ + h * HD_;
  const int vr = tid >> 5;
  const int vc = (tid & 31) * 4;
  const uint32_t vbase_l = lds_addr(&Vt[0][vr][vc]);
  const uint32_t VBUF = (uint32_t)(KCH * KP * 4);

  auto issue = [&](int buf, int k0) {
    async_copy_b128(abase + (uint32_t)buf * ABUF,
                    wrowg + (size_t)ar * T_ + k0 + ac);
#pragma unroll
    for (int i = 0; i < 4; ++i)
      async_copy_b128(vbase_l + (uint32_t)buf * VBUF +
                          (uint32_t)(8 * KP * 4) * i,
                      vhead + (size_t)(k0 + vr + 8 * i) * D_ + vc);
  };

  v8f acc0 = {}, acc1 = {};
  issue(0, 0);
  wait_asynccnt0();
  __syncthreads();

  const int nch = T_ / KCH;  // 64
  for (int ch = 0; ch < nch; ++ch) {
    const int cur = ch & 1;
    if (ch + 1 < nch) issue(cur ^ 1, (ch + 1) * KCH);

    const float* aa = &At[cur][wr * 16 + ln][kL];
    const int nb0 = wc * 32 + ln;
#pragma unroll
    for (int kk = 0; kk < KCH; kk += 4) {
      const v2f a = *(const v2f*)(aa + kk);
      v2f b0, b1;
      b0.x = Vt[cur][kk + kL][nb0];
      b0.y = Vt[cur][kk + kL + 1][nb0];
      b1.x = Vt[cur][kk + kL][nb0 + 16];
      b1.y = Vt[cur][kk + kL + 1][nb0 + 16];
      acc0 = wmma_f32(a, b0, acc0);
      acc1 = wmma_f32(a, b1, acc1);
    }

    if (ch + 1 < nch) wait_asynccnt0();
    __syncthreads();
  }

#pragma unroll
  for (int r = 0; r < 8; ++r) {
    const size_t orow =
        ((size_t)b * T_ + m0 + wr * 16 + r + 8 * hl) * D_ + h * HD_;
    ctx[orow + wc * 32 + ln] = acc0[r];
    ctx[orow + wc * 32 + 16 + ln] = acc1[r];
  }
}

// -------------------------------------------------------------------------
extern "C" void kernel_launch(void* const* d_in, const int* in_sizes, int n_in,
                              void* d_out, int out_size, void* d_ws,
                              size_t ws_size, hipStream_t stream) {
  (void)in_sizes; (void)n_in; (void)out_size; (void)ws_size;
  const float* query = (const float*)d_in[0];
  const float* keys = (const float*)d_in[1];
  const float* values = (const float*)d_in[2];
  const float* Wq = (const float*)d_in[3];
  const float* bq = (const float*)d_in[4];
  const float* Wk = (const float*)d_in[5];
  const float* bk = (const float*)d_in[6];
  const float* Wv = (const float*)d_in[7];
  const float* bv = (const float*)d_in[8];
  const float* Wo = (const float*)d_in[9];
  const float* bo = (const float*)d_in[10];
  const float* relt = (const float*)d_in[11];

  float* out = (float*)d_out;                   // (B,T,D)
  float* weights = out + (size_t)B_ * T_ * D_;  // (B,H,T,T)

  float* ws = (float*)d_ws;
  size_t off = 0;
  float* tq = ws + off;  off += (size_t)B_ * T_ * D_;
  float* tk = ws + off;  off += (size_t)B_ * T_ * D_;
  float* tv = ws + off;  off += (size_t)B_ * T_ * D_;
  float* ctx = ws + off; off += (size_t)B_ * T_ * D_;
  float* proj = ws + off;   off += (size_t)B_ * T_ * 8;
  float* rowmax = ws + off; off += (size_t)B_ * H_ * T_;
  float* rowsum = ws + off; off += (size_t)B_ * H_ * T_;

  const dim3 gblk(256);
  const dim3 ggrid(D_ / BN, MROWS / BM);  // (8, 128)
  gemm_async<<<ggrid, gblk, 0, stream>>>(query, Wq, bq, tq, MROWS, D_, D_);
  gemm_async<<<ggrid, gblk, 0, stream>>>(keys, Wk, bk, tk, MROWS, D_, D_);
  gemm_async<<<ggrid, gblk, 0, stream>>>(values, Wv, bv, tv, MROWS, D_, D_);

  relproj_kernel<<<B_ * T_, 256, 0, stream>>>(tq, relt, proj);

  attn_logits<<<B_ * H_ * (T_ / 64), 256, 0, stream>>>(tq, tk, proj, weights,
                                                       rowmax, rowsum);

  const int nblk = (int)(((size_t)B_ * H_ * T_ * T_ / 4) / 256);
  softmax_norm<<<nblk, 256, 0, stream>>>(weights, rowmax, rowsum);

  attn_ctx<<<B_ * H_ * (T_ / 32), 256, 0, stream>>>(weights, tv, ctx);

  gemm_async<<<ggrid, gblk, 0, stream>>>(ctx, Wo, bo, out, MROWS, D_, D_);
}